// RoIDelta_45715631899302
// MI455X (gfx1250) — compile-verified
//
#include <hip/hip_runtime.h>
#include <math.h>

#define B_    64
#define N_    3000
#define M_    128
#define L_    21
#define CNT_  128
#define MAXV_ 1280   // rand in [1, 1280)

typedef int v4i_ __attribute__((ext_vector_type(4)));

#if __has_builtin(__builtin_amdgcn_global_load_async_to_lds_b128)
#define HAVE_ASYNC_B128 1
#endif
#if __has_builtin(__builtin_amdgcn_s_wait_asynccnt)
#define HAVE_WAIT_ASYNC 1
#endif

// ---------------------------------------------------------------------------
// Kernel 1: zero-fill d_out (bandwidth-bound: ~80.6 MB of stores)
// ---------------------------------------------------------------------------
__global__ void roidelta_zero(float* __restrict__ out, long long n) {
  long long i = (long long)blockIdx.x * blockDim.x + threadIdx.x;
  long long stride = (long long)gridDim.x * blockDim.x;
  long long n4 = n >> 2;
  float4* o4 = (float4*)out;
  for (long long k = i; k < n4; k += stride)
    o4[k] = make_float4(0.f, 0.f, 0.f, 0.f);
  for (long long k = (n4 << 2) + i; k < n; k += stride)
    out[k] = 0.f;
}

// ---------------------------------------------------------------------------
// Kernel 2: per-batch IoU + top-128 selection + sparse scatter.
// One block per batch image. gt boxes/labels staged to LDS via CDNA5
// async global->LDS DMA; selection via LDS histogram (exact argsort-stable
// tie-breaking semantics).
// ---------------------------------------------------------------------------
__global__ __launch_bounds__(1024) void roidelta_main(
    const float* __restrict__ roi,   // (B, N, 4)
    const float* __restrict__ gtb,   // (B, M, 4)
    const int*   __restrict__ gtl,   // (B, M)
    const int*   __restrict__ rp,    // (B, N)
    const int*   __restrict__ rn,    // (B, N)
    float* __restrict__ out)         // deltas (B,N*21,4) then labels (B,N,21)
{
  const int b   = blockIdx.x;
  const int tid = threadIdx.x;
  const int nt  = blockDim.x;

  __shared__ __align__(16) float sGt[M_ * 4];
  __shared__ __align__(16) int   sGl[M_];
  __shared__ int   sMP[N_];
  __shared__ int   sMN[N_];
  __shared__ short sBest[N_];
  __shared__ int   sHP[MAXV_];
  __shared__ int   sHN[MAXV_];
  __shared__ int   sVP, sRP, sVN, sRN;

  // ---- Phase 0: stage gt boxes (2 KB) + labels (512 B) into LDS ----------
  const float* gsrc = gtb + (size_t)b * M_ * 4;
  const int*   lsrc = gtl + (size_t)b * M_;
  if (tid < M_) {
#if defined(HAVE_ASYNC_B128)
    __builtin_amdgcn_global_load_async_to_lds_b128(
        (v4i_*)(gsrc + tid * 4), (v4i_*)(&sGt[tid * 4]), 0, 0);
#else
    ((float4*)sGt)[tid] = ((const float4*)gsrc)[tid];
#endif
  } else if (tid < M_ + M_ / 4) {
    int k = tid - M_;
#if defined(HAVE_ASYNC_B128)
    __builtin_amdgcn_global_load_async_to_lds_b128(
        (v4i_*)(lsrc + k * 4), (v4i_*)(&sGl[k * 4]), 0, 0);
#else
    ((int4*)sGl)[k] = ((const int4*)lsrc)[k];
#endif
  }
#if defined(HAVE_ASYNC_B128)
#if defined(HAVE_WAIT_ASYNC)
  __builtin_amdgcn_s_wait_asynccnt(0);
#else
  asm volatile("s_wait_asynccnt 0" ::: "memory");
#endif
#endif
  for (int v = tid; v < MAXV_; v += nt) { sHP[v] = 0; sHN[v] = 0; }
  __syncthreads();

  // ---- Phase 1: IoU argmax/max per roi, build selection histograms -------
  for (int n = tid; n < N_; n += nt) {
    if (n + nt < N_)
      __builtin_prefetch(roi + ((size_t)b * N_ + n + nt) * 4, 0, 1);
    const float4 rb = *(const float4*)(roi + ((size_t)b * N_ + n) * 4);
    const float r0 = rb.x, r1 = rb.y, r2 = rb.z, r3 = rb.w;
    const float ar = (r2 - r0) * (r3 - r1);
    float best = -1.0f; int bi = 0;
#pragma unroll 4
    for (int m = 0; m < M_; ++m) {
      const float g0 = sGt[m * 4 + 0], g1 = sGt[m * 4 + 1];
      const float g2 = sGt[m * 4 + 2], g3 = sGt[m * 4 + 3];
      const float iy1 = fmaxf(r0, g0), ix1 = fmaxf(r1, g1);
      const float iy2 = fminf(r2, g2), ix2 = fminf(r3, g3);
      const float inter = fmaxf(iy2 - iy1, 0.f) * fmaxf(ix2 - ix1, 0.f);
      const float ag = (g2 - g0) * (g3 - g1);
      const float iou = inter / (ar + ag - inter + 1e-7f);
      if (iou > best) { best = iou; bi = m; }   // first-max like jnp.argmax
    }
    const bool pos = best > 0.5f;
    const bool neg = (best < 0.5f) && (best > 0.1f);
    const int mp = pos ? rp[(size_t)b * N_ + n] : 0;
    const int mn = neg ? rn[(size_t)b * N_ + n] : 0;
    sMP[n] = mp; sMN[n] = mn; sBest[n] = (short)bi;
    if (mp) atomicAdd(&sHP[mp], 1);
    if (mn) atomicAdd(&sHN[mn], 1);
  }
  __syncthreads();

  // ---- Phase 2: find 128th-largest key value v* and residual slots r -----
  if (tid == 0 || tid == 1) {
    const int* h = tid ? sHN : sHP;
    int run = 0, vst = 0, r = 0;
    for (int v = MAXV_ - 1; v >= 1; --v) {
      const int c = h[v];
      if (run + c >= CNT_) { vst = v; r = CNT_ - run; break; }
      run += c;
    }
    if (tid) { sVN = vst; sRN = r; } else { sVP = vst; sRP = r; }
  }
  __syncthreads();
  const int vP = sVP, rPn = sRP, vN = sVN, rNn = sRN;

  float* outL = out + (size_t)B_ * N_ * L_ * 4;   // labels after deltas
  float4* outD4 = (float4*)out;                   // deltas as float4 rows

  // ---- Phase 3: per-roi selection + sparse output writes -----------------
  for (int n = tid; n < N_; n += nt) {
    const int mp = sMP[n], mn = sMN[n];

    bool psel = mp > vP;                 // vP==0 => all masked selected
    if (!psel && vP > 0 && mp == vP) {   // stable tie-break by index
      int ord = 0;
      for (int j = 0; j < n; ++j) ord += (sMP[j] == vP);
      psel = ord < rPn;
    }
    bool nsel = mn > vN;
    if (!nsel && vN > 0 && mn == vN) {
      int ord = 0;
      for (int j = 0; j < n; ++j) ord += (sMN[j] == vN);
      nsel = ord < rNn;
    }

    int lab = -1;
    const int bi = (int)sBest[n];
    if (psel)      lab = sGl[bi];        // 1..20
    else if (nsel) lab = 0;

    if (lab >= 0)
      outL[((size_t)b * N_ + n) * L_ + lab] = 1.0f;

    if (psel) {
      const float4 rb = *(const float4*)(roi + ((size_t)b * N_ + n) * 4);
      const float r0 = rb.x, r1 = rb.y, r2 = rb.z, r3 = rb.w;
      const float g0 = sGt[bi * 4 + 0], g1 = sGt[bi * 4 + 1];
      const float g2 = sGt[bi * 4 + 2], g3 = sGt[bi * 4 + 3];
      float bh = r2 - r0, bw = r3 - r1;
      const float bcy = r0 + 0.5f * bh, bcx = r1 + 0.5f * bw;
      const float gh = g2 - g0, gw = g3 - g1;
      const float gcy = g0 + 0.5f * gh, gcx = g1 + 0.5f * gw;
      bw = (bw == 0.f) ? 1e-3f : bw;
      bh = (bh == 0.f) ? 1e-3f : bh;
      const float dx = (gw == 0.f) ? 0.f : (gcx - bcx) / bw;
      const float dy = (gh == 0.f) ? 0.f : (gcy - bcy) / bh;
      const float dw = (gw == 0.f) ? 0.f : logf(gw / bw);
      const float dh = (gh == 0.f) ? 0.f : logf(gh / bh);
      float4 o;
      o.x = dy / 0.1f; o.y = dx / 0.1f; o.z = dh / 0.2f; o.w = dw / 0.2f;
      outD4[((size_t)b * N_ + n) * L_ + lab] = o;
    }
  }
}

// ---------------------------------------------------------------------------
extern "C" void kernel_launch(void* const* d_in, const int* in_sizes, int n_in,
                              void* d_out, int out_size, void* d_ws, size_t ws_size,
                              hipStream_t stream) {
  (void)in_sizes; (void)n_in; (void)d_ws; (void)ws_size;
  const float* roi = (const float*)d_in[0];
  const float* gtb = (const float*)d_in[1];
  const int*   gtl = (const int*)d_in[2];
  const int*   rp  = (const int*)d_in[3];
  const int*   rn  = (const int*)d_in[4];
  float* out = (float*)d_out;

  const long long n = (long long)out_size;
  long long n4 = n >> 2;
  int blocks = (int)((n4 + 255) / 256);
  if (blocks > 8192) blocks = 8192;
  if (blocks < 1) blocks = 1;
  roidelta_zero<<<blocks, 256, 0, stream>>>(out, n);
  roidelta_main<<<B_, 1024, 0, stream>>>(roi, gtb, gtl, rp, rn, out);
}